// VQVAE_90271622627448
// MI455X (gfx1250) — compile-verified
//
#include <hip/hip_runtime.h>
#include <hip/hip_bf16.h>

typedef float v2f __attribute__((ext_vector_type(2)));
typedef float v8f __attribute__((ext_vector_type(8)));

#define NB    2048
#define HW2   49          // 7*7
#define DLAT  2
#define KCODE 512
#define Q     (NB*HW2)          // 100352
#define OUT_IMG (NB*28*28)      // 1605632
#define BN_EPS 1e-5f

// workspace layout (float offsets)
#define OFF_A      0                         // y1 (6422528 f) then t3 (12845056 f)
#define OFF_ENC    12845056
#define OFF_QUANT  (OFF_ENC   + NB*DLAT*HW2)
#define OFF_T1     (OFF_QUANT + NB*DLAT*HW2)
#define OFF_T2     (OFF_T1    + NB*DLAT*HW2)
#define OFF_STATS  (OFF_T2    + NB*DLAT*HW2)
// offsets inside the stats block
#define ST1   0
#define ST2   32
#define ST3   36
#define ST4   40
#define ST5   44
#define LOSSA 108
#define SC1   112
#define SH1   128
#define SC2   144
#define SH2   146
#define SC3   148
#define SH3   150
#define SC4   152
#define SH4   154
#define SC5   156
#define SH5   188

__global__ __launch_bounds__(256) void k_zero(float* p, int n) {
  int i = blockIdx.x * 256 + threadIdx.x;
  if (i < n) p[i] = 0.f;
}

// BN finalize: mean/var -> scale/shift
__global__ void k_bnfin(const float* __restrict__ stat, const float* __restrict__ g,
                        const float* __restrict__ bb, float* __restrict__ sc,
                        float* __restrict__ sh, int nch, float invcount) {
  int c = threadIdx.x;
  if (c < nch) {
    float mean = stat[c * 2] * invcount;
    float var  = stat[c * 2 + 1] * invcount - mean * mean;
    float s = g[c] * rsqrtf(var + BN_EPS);
    sc[c] = s;
    sh[c] = bb[c] - mean * s;
  }
}

// encoder conv1: 1->16ch, k4 s2 p1, 28x28 -> 14x14, + per-channel stats
__global__ __launch_bounds__(256) void k_conv1(const float* __restrict__ x,
    const float* __restrict__ w, const float* __restrict__ bias,
    float* __restrict__ y1, float* __restrict__ stat) {
  __shared__ float s1[256], s2[256];
  int tid = threadIdx.x;
  int c = blockIdx.y;
  int i = blockIdx.x * 256 + tid;         // 0..401407 (b*196+p)
  int b = i / 196, p = i % 196;
  int y = p / 14, xx = p % 14;
  float acc = bias[c];
#pragma unroll
  for (int ky = 0; ky < 4; ++ky) {
    int iy = 2 * y - 1 + ky;
    if ((unsigned)iy < 28u) {
#pragma unroll
      for (int kx = 0; kx < 4; ++kx) {
        int ix = 2 * xx - 1 + kx;
        if ((unsigned)ix < 28u)
          acc += x[b * 784 + iy * 28 + ix] * w[c * 16 + ky * 4 + kx];
      }
    }
  }
  y1[(b * 16 + c) * 196 + p] = acc;
  s1[tid] = acc; s2[tid] = acc * acc;
  __syncthreads();
  for (int s = 128; s > 0; s >>= 1) {
    if (tid < s) { s1[tid] += s1[tid + s]; s2[tid] += s2[tid + s]; }
    __syncthreads();
  }
  if (tid == 0) { atomicAdd(&stat[c * 2], s1[0]); atomicAdd(&stat[c * 2 + 1], s2[0]); }
}

// encoder conv2: 16->2ch, k4 s2 p1, 14x14 -> 7x7, BN1+ReLU folded on reads, + stats
__global__ __launch_bounds__(256) void k_conv2(const float* __restrict__ y1,
    const float* __restrict__ w, const float* __restrict__ bias,
    const float* __restrict__ stats, float* __restrict__ enc,
    float* __restrict__ stat2) {
  __shared__ float s1[256], s2[256];
  int tid = threadIdx.x;
  int c = blockIdx.y;                     // 0..1
  int i = blockIdx.x * 256 + tid;         // 0..100351
  int b = i / 49, p = i % 49;
  int y = p / 7, xx = p % 7;
  float acc = bias[c];
  for (int ic = 0; ic < 16; ++ic) {
    float sc = stats[SC1 + ic], sh = stats[SH1 + ic];
    const float* yb = y1 + (b * 16 + ic) * 196;
#pragma unroll
    for (int ky = 0; ky < 4; ++ky) {
      int iy = 2 * y - 1 + ky;
      if ((unsigned)iy < 14u) {
#pragma unroll
        for (int kx = 0; kx < 4; ++kx) {
          int ix = 2 * xx - 1 + kx;
          if ((unsigned)ix < 14u) {
            float v = fmaxf(yb[iy * 14 + ix] * sc + sh, 0.f);
            acc += v * w[((c * 16 + ic) * 4 + ky) * 4 + kx];
          }
        }
      }
    }
  }
  enc[(b * 2 + c) * 49 + p] = acc;
  s1[tid] = acc; s2[tid] = acc * acc;
  __syncthreads();
  for (int s = 128; s > 0; s >>= 1) {
    if (tid < s) { s1[tid] += s1[tid + s]; s2[tid] += s2[tid + s]; }
    __syncthreads();
  }
  if (tid == 0) { atomicAdd(&stat2[c * 2], s1[0]); atomicAdd(&stat2[c * 2 + 1], s2[0]); }
}

// VQ: WMMA f32 16x16x4 distance GEMM + argmin + quantize + loss accumulation.
// Codebook (4 KB) staged into LDS via CDNA5 async global->LDS copy.
// One wave handles 16 queries x 512 codes (32 N-tiles).
__global__ __launch_bounds__(256) void k_vq(const float* __restrict__ enc,
    const float* __restrict__ embed, const float* __restrict__ stats,
    float* __restrict__ quant, float* __restrict__ closest_out,
    float* __restrict__ lossAcc) {
  __shared__ float lembed[KCODE * 2];     // 512 x 2 floats = 4 KB
  int tid  = threadIdx.x;

  // Async stage: 256 threads x 16 B = whole table; ASYNCcnt tracked.
  {
    unsigned lds_b = (unsigned)(unsigned long long)(const void*)(&lembed[0])
                     + (unsigned)tid * 16u;
    const float* gsrc = embed + tid * 4;
    asm volatile("global_load_async_to_lds_b128 %0, %1, off"
                 :: "v"(lds_b), "v"(gsrc) : "memory");
    asm volatile("s_wait_asynccnt 0" ::: "memory");
  }
  __syncthreads();

  const float sA = stats[SC2 + 0], sB = stats[SC2 + 1];
  const float hA = stats[SH2 + 0], hB = stats[SH2 + 1];
  int lane = tid & 31;
  int wid  = tid >> 5;
  int tile = blockIdx.x * 8 + wid;        // 0..6271
  int m0 = tile * 16;
  int j = lane & 15;
  int g = lane >> 4;

  // A matrix: lanes 0-15 hold row m=j with K0,K1; lanes 16-31 hold K2,K3 (zero, K=2)
  float a0 = 0.f, a1 = 0.f;
  if (g == 0) {
    int m = m0 + j;
    int b = m / HW2, n = m % HW2;
    int f0 = b * (DLAT * HW2) + n * 2;    // raw NCHW-flat view (B, 49, 2)
    float r0 = enc[f0], r1 = enc[f0 + 1];
    int ch0 = (n * 2) / HW2;              // BN channel of each interleaved element
    int ch1 = (n * 2 + 1) / HW2;
    float q0 = r0 * (ch0 ? sB : sA) + (ch0 ? hB : hA);
    float q1 = r1 * (ch1 ? sB : sA) + (ch1 ? hB : hA);
    a0 = fmaxf(q0, 0.f);
    a1 = fmaxf(q1, 0.f);
  }
  v2f A; A[0] = a0; A[1] = a1;

  float minv[8];
  int   mini[8];
#pragma unroll
  for (int r = 0; r < 8; ++r) { minv[r] = 3.4e38f; mini[r] = 0; }

#pragma unroll 4
  for (int t = 0; t < KCODE / 16; ++t) {
    int nn = t * 16 + j;                  // this lane's column (code index)
    float e0 = lembed[nn * 2], e1 = lembed[nn * 2 + 1];
    float ee = e0 * e0 + e1 * e1;
    v2f Bm; Bm[0] = (g == 0) ? e0 : 0.f; Bm[1] = (g == 0) ? e1 : 0.f;
    v8f C = {};
    C = __builtin_amdgcn_wmma_f32_16x16x4_f32(false, A, false, Bm, (short)0, C,
                                              false, false);
#pragma unroll
    for (int r = 0; r < 8; ++r) {
      float cand = ee - 2.0f * C[r];      // ||q||^2 is row-constant: argmin-invariant
      // nn strictly increases within a lane -> strict < keeps the first min
      if (cand < minv[r]) { minv[r] = cand; mini[r] = nn; }
    }
  }

  // min+index reduce across the 16 lanes of each half (xor<16 stays in-half)
#pragma unroll
  for (int r = 0; r < 8; ++r) {
#pragma unroll
    for (int off = 8; off >= 1; off >>= 1) {
      float ov = __shfl_xor(minv[r], off);
      int   oi = __shfl_xor(mini[r], off);
      if (ov < minv[r] || (ov == minv[r] && oi < mini[r])) {
        minv[r] = ov; mini[r] = oi;
      }
    }
  }

  float rowloss = 0.f;
#pragma unroll
  for (int r = 0; r < 8; ++r) {
    int row = r + 8 * g;                  // C row covered by this half at VGPR r
    float q0r = __shfl(a0, row);          // q lives in lanes 0..15 (all active)
    float q1r = __shfl(a1, row);
    if (j == r) {                         // one writer lane per row
      int m = m0 + row;
      int best = mini[r];
      float eb0 = lembed[best * 2], eb1 = lembed[best * 2 + 1];
      int b = m / HW2, n = m % HW2;
      int f0 = b * (DLAT * HW2) + n * 2;
      quant[f0]     = eb0;                // straight-through forward = quantized
      quant[f0 + 1] = eb1;
      closest_out[m] = (float)best;
      float d0 = eb0 - q0r, d1 = eb1 - q1r;
      rowloss = d0 * d0 + d1 * d1;        // exact ||e-q||^2 for the loss
    }
  }
  // wave then block reduce loss -> one atomic per block
#pragma unroll
  for (int off = 16; off >= 1; off >>= 1) rowloss += __shfl_xor(rowloss, off);
  __shared__ float ls[8];
  if (lane == 0) ls[wid] = rowloss;
  __syncthreads();
  if (tid == 0) {
    float s = 0.f;
    for (int w = 0; w < 8; ++w) s += ls[w];
    atomicAdd(lossAcc, s);
  }
}

// pointwise 1x1 conv on 2ch, optional folded BN+ReLU on input, + stats
__global__ __launch_bounds__(256) void k_pw(const float* __restrict__ in,
    const float* __restrict__ w, const float* __restrict__ bias,
    const float* __restrict__ scp, const float* __restrict__ shp, int applyBn,
    float* __restrict__ out, float* __restrict__ stat) {
  __shared__ float s1[256], s2[256];
  int tid = threadIdx.x;
  int o = blockIdx.y;
  int i = blockIdx.x * 256 + tid;
  int b = i / 49, p = i % 49;
  float v0 = in[b * 98 + p];
  float v1 = in[b * 98 + 49 + p];
  if (applyBn) {
    v0 = fmaxf(v0 * scp[0] + shp[0], 0.f);
    v1 = fmaxf(v1 * scp[1] + shp[1], 0.f);
  }
  float acc = bias[o] + w[o * 2] * v0 + w[o * 2 + 1] * v1;
  out[b * 98 + o * 49 + p] = acc;
  s1[tid] = acc; s2[tid] = acc * acc;
  __syncthreads();
  for (int s = 128; s > 0; s >>= 1) {
    if (tid < s) { s1[tid] += s1[tid + s]; s2[tid] += s2[tid + s]; }
    __syncthreads();
  }
  if (tid == 0) { atomicAdd(&stat[o * 2], s1[0]); atomicAdd(&stat[o * 2 + 1], s2[0]); }
}

// convT1: 2->32ch, k4 s2 p1, 7x7 -> 14x14, BN4+ReLU folded on reads, + stats
__global__ __launch_bounds__(256) void k_convt1(const float* __restrict__ t2,
    const float* __restrict__ w, const float* __restrict__ bias,
    const float* __restrict__ stats, float* __restrict__ t3,
    float* __restrict__ stat5) {
  __shared__ float s1[256], s2[256];
  int tid = threadIdx.x;
  int o = blockIdx.y;                     // 0..31
  int i = blockIdx.x * 256 + tid;         // 0..401407
  int b = i / 196, p = i % 196;
  int y = p / 14, xx = p % 14;
  float sc0 = stats[SC4 + 0], sh0 = stats[SH4 + 0];
  float sc1 = stats[SC4 + 1], sh1 = stats[SH4 + 1];
  float acc = bias[o];
#pragma unroll
  for (int ky = 0; ky < 4; ++ky) {
    int t = y + 1 - ky;
    if (t >= 0 && !(t & 1)) {
      int iy = t >> 1;
      if (iy < 7) {
#pragma unroll
        for (int kx = 0; kx < 4; ++kx) {
          int u = xx + 1 - kx;
          if (u >= 0 && !(u & 1)) {
            int ix = u >> 1;
            if (ix < 7) {
              float h0 = fmaxf(t2[b * 98 + iy * 7 + ix] * sc0 + sh0, 0.f);
              float h1 = fmaxf(t2[b * 98 + 49 + iy * 7 + ix] * sc1 + sh1, 0.f);
              acc += h0 * w[(o * 4 + ky) * 4 + kx]                 // in-ch 0
                   + h1 * w[((32 + o) * 4 + ky) * 4 + kx];          // in-ch 1
            }
          }
        }
      }
    }
  }
  t3[(b * 32 + o) * 196 + p] = acc;
  s1[tid] = acc; s2[tid] = acc * acc;
  __syncthreads();
  for (int s = 128; s > 0; s >>= 1) {
    if (tid < s) { s1[tid] += s1[tid + s]; s2[tid] += s2[tid + s]; }
    __syncthreads();
  }
  if (tid == 0) { atomicAdd(&stat5[o * 2], s1[0]); atomicAdd(&stat5[o * 2 + 1], s2[0]); }
}

// convT2: 32->1ch, k4 s2 p1, 14x14 -> 28x28, BN5+ReLU folded, sigmoid
__global__ __launch_bounds__(256) void k_convt2(const float* __restrict__ t3,
    const float* __restrict__ w, const float* __restrict__ bias,
    const float* __restrict__ stats, float* __restrict__ outimg) {
  int i = blockIdx.x * 256 + threadIdx.x; // 0..1605631
  int b = i / 784, p = i % 784;
  int y = p / 28, xx = p % 28;
  float acc = bias[0];
#pragma unroll
  for (int ky = 0; ky < 4; ++ky) {
    int t = y + 1 - ky;
    if (t >= 0 && !(t & 1)) {
      int iy = t >> 1;
      if (iy < 14) {
#pragma unroll
        for (int kx = 0; kx < 4; ++kx) {
          int u = xx + 1 - kx;
          if (u >= 0 && !(u & 1)) {
            int ix = u >> 1;
            if (ix < 14) {
              const float* t3b = t3 + b * 32 * 196 + iy * 14 + ix;
              for (int o = 0; o < 32; ++o) {
                float hv = fmaxf(t3b[o * 196] * stats[SC5 + o] + stats[SH5 + o], 0.f);
                acc += hv * w[o * 16 + ky * 4 + kx];
              }
            }
          }
        }
      }
    }
  }
  outimg[i] = 1.f / (1.f + __expf(-acc));
}

__global__ void k_loss(const float* __restrict__ lossAcc, float* __restrict__ out) {
  if (threadIdx.x == 0)
    out[0] = 1.2f * lossAcc[0] / (float)(Q * DLAT);  // codebook + 0.2*commitment
}

extern "C" void kernel_launch(void* const* d_in, const int* in_sizes, int n_in,
                              void* d_out, int out_size, void* d_ws, size_t ws_size,
                              hipStream_t stream) {
  (void)in_sizes; (void)n_in; (void)out_size; (void)ws_size;
  const float* x    = (const float*)d_in[0];
  const float* ew1  = (const float*)d_in[1];
  const float* eb1  = (const float*)d_in[2];
  const float* g1   = (const float*)d_in[3];
  const float* b1   = (const float*)d_in[4];
  const float* ew2  = (const float*)d_in[5];
  const float* eb2  = (const float*)d_in[6];
  const float* g2   = (const float*)d_in[7];
  const float* b2   = (const float*)d_in[8];
  const float* emb  = (const float*)d_in[9];
  const float* dw1  = (const float*)d_in[10];
  const float* dbi1 = (const float*)d_in[11];
  const float* dg1  = (const float*)d_in[12];
  const float* dbb1 = (const float*)d_in[13];
  const float* dw2  = (const float*)d_in[14];
  const float* dbi2 = (const float*)d_in[15];
  const float* dg2  = (const float*)d_in[16];
  const float* dbb2 = (const float*)d_in[17];
  const float* tw1  = (const float*)d_in[18];
  const float* tbi1 = (const float*)d_in[19];
  const float* dg3  = (const float*)d_in[20];
  const float* dbb3 = (const float*)d_in[21];
  const float* tw2  = (const float*)d_in[22];
  const float* tbi2 = (const float*)d_in[23];

  float* ws    = (float*)d_ws;
  float* y1    = ws + OFF_A;      // encoder activation
  float* t3    = ws + OFF_A;      // reused after y1 is consumed
  float* enc   = ws + OFF_ENC;
  float* quant = ws + OFF_QUANT;
  float* t1    = ws + OFF_T1;
  float* t2    = ws + OFF_T2;
  float* st    = ws + OFF_STATS;

  float* outimg  = (float*)d_out;
  float* closest = outimg + OUT_IMG;
  float* lossout = closest + Q;

  k_zero<<<1, 256, 0, stream>>>(st, 256);
  k_conv1<<<dim3(1568, 16), 256, 0, stream>>>(x, ew1, eb1, y1, st + ST1);
  k_bnfin<<<1, 32, 0, stream>>>(st + ST1, g1, b1, st + SC1, st + SH1, 16,
                                1.f / (2048.f * 196.f));
  k_conv2<<<dim3(392, 2), 256, 0, stream>>>(y1, ew2, eb2, st, enc, st + ST2);
  k_bnfin<<<1, 32, 0, stream>>>(st + ST2, g2, b2, st + SC2, st + SH2, 2,
                                1.f / (2048.f * 49.f));
  k_vq<<<784, 256, 0, stream>>>(enc, emb, st, quant, closest, st + LOSSA);
  k_pw<<<dim3(392, 2), 256, 0, stream>>>(quant, dw1, dbi1, nullptr, nullptr, 0,
                                         t1, st + ST3);
  k_bnfin<<<1, 32, 0, stream>>>(st + ST3, dg1, dbb1, st + SC3, st + SH3, 2,
                                1.f / (2048.f * 49.f));
  k_pw<<<dim3(392, 2), 256, 0, stream>>>(t1, dw2, dbi2, st + SC3, st + SH3, 1,
                                         t2, st + ST4);
  k_bnfin<<<1, 32, 0, stream>>>(st + ST4, dg2, dbb2, st + SC4, st + SH4, 2,
                                1.f / (2048.f * 49.f));
  k_convt1<<<dim3(1568, 32), 256, 0, stream>>>(t2, tw1, tbi1, st, t3, st + ST5);
  k_bnfin<<<1, 32, 0, stream>>>(st + ST5, dg3, dbb3, st + SC5, st + SH5, 32,
                                1.f / (2048.f * 196.f));
  k_convt2<<<6272, 256, 0, stream>>>(t3, tw2, tbi2, st, outimg);
  k_loss<<<1, 1, 0, stream>>>(st + LOSSA, lossout);
}